// NGCF_80350248174011
// MI455X (gfx1250) — compile-verified
//
#include <hip/hip_runtime.h>
#include <hip/hip_bf16.h>

#define N_USERS 50000
#define N_ITEMS 75000
#define N_NODES (N_USERS + N_ITEMS)
#define NNZ     1250000
#define EMB     64
#define N_LAYERS 3
#define BATCH   4096
#define EPSV    1e-12f

typedef __attribute__((ext_vector_type(2)))  float    v2f;
typedef __attribute__((ext_vector_type(8)))  float    v8f;

// ---------------------------------------------------------------------------
// SpMM scatter: agg[row[e], :] += vals[e] * node_feat[col[e], :]
// One thread per (edge, 2-float chunk): 32 threads cover the 64-wide feature.
// Edge-list stream is sequential -> prefetch it ahead; feature gather is random.
// ---------------------------------------------------------------------------
__global__ void ngcf_spmm(const int* __restrict__ row, const int* __restrict__ col,
                          const float* __restrict__ vals, const float* __restrict__ nf,
                          float* __restrict__ agg) {
  long long tid = (long long)blockIdx.x * blockDim.x + threadIdx.x;
  const long long total = (long long)NNZ * 32;
  if (tid >= total) return;
  int e = (int)(tid >> 5);
  int c = ((int)tid & 31) * 2;
  if (((int)tid & 31) == 0) {
    int ep = e + 4096; if (ep > NNZ - 1) ep = NNZ - 1;
    __builtin_prefetch(col + ep, 0, 0);
    __builtin_prefetch(row + ep, 0, 0);
    __builtin_prefetch(vals + ep, 0, 0);
  }
  float v = vals[e];
  int src = col[e];
  int dst = row[e];
  float2 f = *(const float2*)(nf + (long long)src * EMB + c);
  atomicAdd(agg + (long long)dst * EMB + c,     v * f.x);
  atomicAdd(agg + (long long)dst * EMB + c + 1, v * f.y);
}

// ---------------------------------------------------------------------------
// Dense layer: nf_out = leaky(agg@W1^T + b1) + leaky((agg*nf_in)@W2^T + b2)
// 256 threads = 8 waves; weights staged in LDS once per block; one wave32 per
// 16-node tile; full-f32 V_WMMA_F32_16X16X4_F32 accumulation (K=64 -> 16 steps).
// ---------------------------------------------------------------------------
__global__ void __launch_bounds__(256)
ngcf_dense_wmma(const float* __restrict__ agg, const float* __restrict__ nf_in,
                const float* __restrict__ W1, const float* __restrict__ b1,
                const float* __restrict__ W2, const float* __restrict__ b2,
                float* __restrict__ nf_out) {
  __shared__ float w1s[EMB * EMB];   // 16 KB
  __shared__ float w2s[EMB * EMB];   // 16 KB

  // Cooperative weight staging: 4096 floats/array, 256 threads, float4 chunks.
  const int t = threadIdx.x;
#pragma unroll
  for (int i = 0; i < 4; ++i) {
    int idx = (t + i * 256) * 4;
    *(float4*)(w1s + idx) = *(const float4*)(W1 + idx);
    *(float4*)(w2s + idx) = *(const float4*)(W2 + idx);
  }
  __syncthreads();                    // all waves reach the barrier (exit below)

  const int wave = t >> 5;            // 0..7
  const int lane = t & 31;
  const int r = lane & 15;            // A-row index / D column index
  const int h = lane >> 4;            // lane half
  const int node_base = (blockIdx.x * 8 + wave) * 16;
  if (node_base >= N_NODES) return;   // wave-uniform: EXEC all-1s for WMMA

  // A-row pointers (clamped for the tail tile; stores predicated below)
  int nrow = node_base + r;
  if (nrow >= N_NODES) nrow = N_NODES - 1;
  const float* aggp = agg   + (long long)nrow * EMB;
  const float* nfp  = nf_in + (long long)nrow * EMB;

#pragma unroll
  for (int jt = 0; jt < 4; ++jt) {
    const int colr = jt * 16 + r;     // output column handled by this lane
    const float* w1p = w1s + colr * EMB;
    const float* w2p = w2s + colr * EMB;
    v8f c1 = {}; v8f c2 = {};

#pragma unroll
    for (int ks = 0; ks < 16; ++ks) {
      int k = ks * 4 + 2 * h;         // ISA A-layout: lane half selects K pair
      float2 a  = *(const float2*)(aggp + k);     // global (L0-resident)
      float2 f  = *(const float2*)(nfp  + k);
      float2 w1v = *(const float2*)(w1p + k);     // LDS (ds_load_b64)
      float2 w2v = *(const float2*)(w2p + k);
      v2f av;  av.x  = a.x;        av.y  = a.y;
      v2f pv;  pv.x  = a.x * f.x;  pv.y  = a.y * f.y;
      v2f b1v; b1v.x = w1v.x;      b1v.y = w1v.y;
      v2f b2v; b2v.x = w2v.x;      b2v.y = w2v.y;
      c1 = __builtin_amdgcn_wmma_f32_16x16x4_f32(false, av, false, b1v,
                                                 (short)0, c1, false, false);
      c2 = __builtin_amdgcn_wmma_f32_16x16x4_f32(false, pv, false, b2v,
                                                 (short)0, c2, false, false);
    }

    const float bias1 = b1[colr];
    const float bias2 = b2[colr];
#pragma unroll
    for (int i = 0; i < 8; ++i) {
      int node = node_base + i + 8 * h;          // D layout: VGPR i -> M = i + 8h
      float x1 = c1[i] + bias1; x1 = fmaxf(x1, 0.2f * x1);  // leaky_relu(.,0.2)
      float x2 = c2[i] + bias2; x2 = fmaxf(x2, 0.2f * x2);
      if (node < N_NODES) nf_out[(long long)node * EMB + colr] = x1 + x2;
    }
  }
}

// ---------------------------------------------------------------------------
// Gather one 64-wide segment of the concat embedding into d_out for the
// 3*BATCH batch rows; optional L2 normalization via wave32 shuffle reduction.
// ---------------------------------------------------------------------------
__global__ void ngcf_gather_seg(const int* __restrict__ user, const int* __restrict__ pos,
                                const int* __restrict__ neg, const float* __restrict__ src,
                                int seg, int do_norm, float* __restrict__ out) {
  int rowi = blockIdx.x * 4 + (threadIdx.x >> 5);
  if (rowi >= 3 * BATCH) return;
  int lane = threadIdx.x & 31;
  int node;
  if (rowi < BATCH)              node = user[rowi];
  else if (rowi < 2 * BATCH)     node = N_USERS + pos[rowi - BATCH];
  else                           node = N_USERS + neg[rowi - 2 * BATCH];

  float2 f = *(const float2*)(src + (long long)node * EMB + lane * 2);
  float scale = 1.0f;
  if (do_norm) {
    float s = f.x * f.x + f.y * f.y;
#pragma unroll
    for (int off = 16; off > 0; off >>= 1) s += __shfl_xor(s, off, 32);
    scale = 1.0f / fmaxf(sqrtf(s), EPSV);
  }
  float* o = out + (long long)rowi * (EMB * (N_LAYERS + 1)) + seg * EMB + lane * 2;
  o[0] = f.x * scale;
  o[1] = f.y * scale;
}

// ---------------------------------------------------------------------------
extern "C" void kernel_launch(void* const* d_in, const int* in_sizes, int n_in,
                              void* d_out, int out_size, void* d_ws, size_t ws_size,
                              hipStream_t stream) {
  (void)in_sizes; (void)n_in; (void)out_size; (void)ws_size;
  const int*   user = (const int*)d_in[0];
  const int*   pos  = (const int*)d_in[1];
  const int*   neg  = (const int*)d_in[2];
  const int*   row  = (const int*)d_in[3];
  const int*   col  = (const int*)d_in[4];
  const float* vals = (const float*)d_in[5];
  const float* emb  = (const float*)d_in[6];
  const float* W1   = (const float*)d_in[7];
  const float* b1   = (const float*)d_in[8];
  const float* W2   = (const float*)d_in[9];
  const float* b2   = (const float*)d_in[10];
  float* out = (float*)d_out;

  const size_t FEAT_BYTES = (size_t)N_NODES * EMB * sizeof(float);  // 32 MB
  float* agg = (float*)d_ws;
  float* nfA = (float*)((char*)d_ws + FEAT_BYTES);
  float* nfB = (float*)((char*)d_ws + 2 * FEAT_BYTES);

  const long long spmm_threads = (long long)NNZ * 32;
  const int spmm_blocks   = (int)((spmm_threads + 255) / 256);
  const int dense_tiles   = (N_NODES + 15) / 16;
  const int dense_blocks  = (dense_tiles + 7) / 8;
  const int gather_blocks = (3 * BATCH + 3) / 4;

  // Segment 0: raw embeddings (no normalization)
  ngcf_gather_seg<<<gather_blocks, 128, 0, stream>>>(user, pos, neg, emb, 0, 0, out);

  const float* nf_cur = emb;
  for (int l = 0; l < N_LAYERS; ++l) {
    float* nf_next = (l & 1) ? nfB : nfA;
    hipMemsetAsync(agg, 0, FEAT_BYTES, stream);
    ngcf_spmm<<<spmm_blocks, 256, 0, stream>>>(row, col, vals, nf_cur, agg);
    ngcf_dense_wmma<<<dense_blocks, 256, 0, stream>>>(
        agg, nf_cur, W1 + l * EMB * EMB, b1 + l * EMB,
        W2 + l * EMB * EMB, b2 + l * EMB, nf_next);
    ngcf_gather_seg<<<gather_blocks, 128, 0, stream>>>(user, pos, neg, nf_next,
                                                       l + 1, 1, out);
    nf_cur = nf_next;
  }
}